// GCNConv_32487132627452
// MI455X (gfx1250) — compile-verified
//
#include <hip/hip_runtime.h>
#include <hip/hip_bf16.h>

#define DIM 128

typedef float v2f __attribute__((ext_vector_type(2)));
typedef float v8f __attribute__((ext_vector_type(8)));

__device__ __forceinline__ void atomAddF(float* p, float v) {
    __hip_atomic_fetch_add(p, v, __ATOMIC_RELAXED, __HIP_MEMORY_SCOPE_AGENT);
}

// ---------------------------------------------------------------- deg kernels
__global__ void deg_init_kernel(float* __restrict__ deg, int n) {
    int i = blockIdx.x * blockDim.x + threadIdx.x;
    if (i < n) deg[i] = 1.0f;               // self-loop contributes 1 to degree
}

__global__ void deg_count_kernel(const int* __restrict__ ei, float* __restrict__ deg, int E) {
    int e = blockIdx.x * blockDim.x + threadIdx.x;
    if (e < E) atomAddF(&deg[ei[E + e]], 1.0f);   // dst row = edge_index[1][e]
}

__global__ void deg_to_dis_kernel(float* __restrict__ deg, int n) {
    int i = blockIdx.x * blockDim.x + threadIdx.x;
    if (i < n) {
        float d = deg[i];
        deg[i] = (d > 0.0f) ? rsqrtf(d) : 0.0f;   // deg >= 1 always, but match ref
    }
}

// ------------------------------------------------- self-loop: agg = x * dis^2
// One wave32 per row; lane handles 4 consecutive cols (float4, coalesced 512B).
__global__ __launch_bounds__(256) void self_loop_kernel(const float* __restrict__ x,
                                                        const float* __restrict__ dis,
                                                        float* __restrict__ agg, int n) {
    int t = blockIdx.x * blockDim.x + threadIdx.x;
    int row = t >> 5;
    int lane = t & 31;
    if (row >= n) return;
    float s = dis[row];
    s = s * s;
    const float4* xr = reinterpret_cast<const float4*>(x + (size_t)row * DIM);
    float4* ar = reinterpret_cast<float4*>(agg + (size_t)row * DIM);
    float4 v = xr[lane];
    v.x *= s; v.y *= s; v.z *= s; v.w *= s;
    ar[lane] = v;
}

// ------------------------------------- edge scatter: agg[dst] += x[src]*norm
// One wave32 per edge: b128 gather of x[src], 4 f32 atomics per lane to agg[dst].
__global__ __launch_bounds__(256) void edge_scatter_kernel(const float* __restrict__ x,
                                                           const float* __restrict__ dis,
                                                           const int* __restrict__ ei,
                                                           float* __restrict__ agg, int E) {
    int t = blockIdx.x * blockDim.x + threadIdx.x;
    int e = t >> 5;
    int lane = t & 31;
    if (e >= E) return;
    int s = ei[e];          // edge_index[0][e] : message source
    int d = ei[E + e];      // edge_index[1][e] : aggregation target
    float nrm = dis[s] * dis[d];
    const float4* xr = reinterpret_cast<const float4*>(x + (size_t)s * DIM);
    float4 v = xr[lane];
    float* dst = agg + (size_t)d * DIM + lane * 4;
    atomAddF(dst + 0, v.x * nrm);
    atomAddF(dst + 1, v.y * nrm);
    atomAddF(dst + 2, v.z * nrm);
    atomAddF(dst + 3, v.w * nrm);
}

// -------------------------------------------------- GEMM: out = agg @ W (f32)
// Block = 16-row strip of agg across all 128 output cols. 8 waves, wave w owns
// the 16x16 tile at cols [16w,16w+16). A strip + W K-chunks staged in LDS with
// padded strides (132 / 136 floats) to avoid 64-bank conflicts.
// V_WMMA_F32_16X16X4_F32 VGPR layout (ISA 7.12.2):
//   A (16x4):  lane l -> M=l&15;  VGPR0 = K=2*(l>>4), VGPR1 = K=2*(l>>4)+1
//   B (4x16):  lane l -> N=l&15;  VGPR0 = K=2*(l>>4), VGPR1 = K=2*(l>>4)+1
//   C/D:       lane l, VGPR v -> D[v + 8*(l>>4)][l&15]
#define LDA 132
#define LDW 136
#define KCHUNK 32

__global__ __launch_bounds__(256) void gemm_wmma_kernel(const float* __restrict__ A,
                                                        const float* __restrict__ W,
                                                        float* __restrict__ out, int n) {
    __shared__ float lA[16 * LDA];            //  8448 B
    __shared__ float lW[KCHUNK * LDW];        // 17408 B
    const int tid = threadIdx.x;
    const int wave = tid >> 5;
    const int lane = tid & 31;
    const int rowBase = blockIdx.x * 16;

    // Stage the 16x128 A strip (coalesced; 8 elements per thread).
    for (int i = tid; i < 16 * DIM; i += 256) {
        int r = i >> 7, c = i & 127;
        int gr = rowBase + r;
        lA[r * LDA + c] = (gr < n) ? A[(size_t)gr * DIM + c] : 0.0f;
    }

    const int m = lane & 15;          // M (for A) / N (for B) index
    const int hi = lane >> 4;         // K sub-phase
    const int colBase = wave * 16;    // this wave's output-column tile

    v8f acc = {0.f, 0.f, 0.f, 0.f, 0.f, 0.f, 0.f, 0.f};

    for (int kc = 0; kc < DIM; kc += KCHUNK) {
        __syncthreads();
        // Stage W rows [kc, kc+KCHUNK) (coalesced; 16 elements per thread).
        for (int i = tid; i < KCHUNK * DIM; i += 256) {
            int r = i >> 7, c = i & 127;
            lW[r * LDW + c] = W[(size_t)(kc + r) * DIM + c];
        }
        __syncthreads();

#pragma unroll
        for (int k = 0; k < KCHUNK; k += 4) {
            v2f a, b;
            const float* ap = &lA[m * LDA + kc + k + 2 * hi];
            a.x = ap[0];
            a.y = ap[1];
            b.x = lW[(k + 2 * hi) * LDW + colBase + m];
            b.y = lW[(k + 2 * hi + 1) * LDW + colBase + m];
            acc = __builtin_amdgcn_wmma_f32_16x16x4_f32(
                /*neg_a=*/false, a, /*neg_b=*/false, b,
                /*c_mod=*/(short)0, acc, /*reuse_a=*/false, /*reuse_b=*/false);
        }
    }

    // Write the 16x16 f32 tile per C/D layout.
#pragma unroll
    for (int v = 0; v < 8; ++v) {
        int gr = rowBase + v + 8 * hi;
        if (gr < n) out[(size_t)gr * DIM + colBase + m] = acc[v];
    }
}

// ----------------------------------------------------------------- launcher
extern "C" void kernel_launch(void* const* d_in, const int* in_sizes, int n_in,
                              void* d_out, int out_size, void* d_ws, size_t ws_size,
                              hipStream_t stream) {
    const float* x = (const float*)d_in[0];   // [N,128] f32
    const int*   ei = (const int*)d_in[1];    // [2,E] int
    const float* W = (const float*)d_in[2];   // [128,128] f32
    float* out = (float*)d_out;               // [N,128] f32

    const int N = in_sizes[0] / DIM;
    const int E = in_sizes[1] / 2;

    float* deg = (float*)d_ws;                                  // N floats (becomes dis)
    size_t aggOff = (((size_t)N * sizeof(float)) + 255) & ~(size_t)255;
    float* agg = (float*)((char*)d_ws + aggOff);                // N*128 floats

    deg_init_kernel<<<(N + 255) / 256, 256, 0, stream>>>(deg, N);
    deg_count_kernel<<<(E + 255) / 256, 256, 0, stream>>>(ei, deg, E);
    deg_to_dis_kernel<<<(N + 255) / 256, 256, 0, stream>>>(deg, N);
    self_loop_kernel<<<(N + 7) / 8, 256, 0, stream>>>(x, deg, agg, N);
    edge_scatter_kernel<<<(E + 7) / 8, 256, 0, stream>>>(x, deg, ei, agg, E);
    gemm_wmma_kernel<<<(N + 15) / 16, 256, 0, stream>>>(agg, W, out, N);
}